// MonteCarloGCN_37744172597765
// MI455X (gfx1250) — compile-verified
//
#include <hip/hip_runtime.h>

#define N_NODES 100000
#define N_EDGES 3200000
#define F_IN    128
#define HID     16
#define N_DRUGS 2000

typedef __attribute__((ext_vector_type(2))) float v2f;
typedef __attribute__((ext_vector_type(8))) float v8f;

// Hardware fp32 atomic (global_atomic_add_f32), no CAS fallback.
__device__ __forceinline__ void atomic_fadd_hw(float* p, float v) {
#if defined(__HIP_DEVICE_COMPILE__)
    unsafeAtomicAdd(p, v);
#else
    atomicAdd(p, v);
#endif
}

// ---------------- degree / normalization ----------------

__global__ void k_deg_init(float* __restrict__ deg) {
    int i = blockIdx.x * blockDim.x + threadIdx.x;
    if (i < N_NODES) deg[i] = 1.0f;  // self-loop contributes 1 to every node
}

__global__ void k_deg_accum(const int* __restrict__ ei, float* __restrict__ deg) {
    int e = blockIdx.x * blockDim.x + threadIdx.x;
    if (e < N_EDGES) atomic_fadd_hw(&deg[ei[N_EDGES + e]], 1.0f);  // dst row
}

__global__ void k_rsqrt(float* __restrict__ deg) {
    int i = blockIdx.x * blockDim.x + threadIdx.x;
    if (i < N_NODES) deg[i] = rsqrtf(deg[i]);  // deg >= 1 always (self-loops)
}

// ---------------- WMMA GEMM: [M x K] @ [K x 16] -> [M x 16] ----------------
// One wave per 16-row tile; 8 waves per block. fp32 WMMA 16x16x4 keeps full
// reference precision. EXEC stays all-ones inside each live wave (tile guard
// is wave-uniform).

template <int K, bool RELU>
__global__ void k_gemm16(const float* __restrict__ A, const float* __restrict__ B,
                         float* __restrict__ C, int ntiles) {
    int wave = threadIdx.x >> 5;
    int lane = threadIdx.x & 31;
    int tile = blockIdx.x * 8 + wave;
    if (tile >= ntiles) return;               // whole-wave uniform exit

    int half = lane >> 4;                     // 0: K={k0,k0+1}, 1: K={k0+2,k0+3}
    int col  = lane & 15;
    int row  = tile * 16 + (lane & 15);       // A: M = lane&15 per ISA layout

    v8f c = {};
#pragma unroll
    for (int k0 = 0; k0 < K; k0 += 4) {
        int ka = k0 + half * 2;
        float a0 = A[row * K + ka];
        float a1 = A[row * K + ka + 1];
        if (RELU) { a0 = fmaxf(a0, 0.0f); a1 = fmaxf(a1, 0.0f); }
        v2f a, b;
        a.x = a0;                 a.y = a1;
        b.x = B[ka * HID + col];  b.y = B[(ka + 1) * HID + col];
        c = __builtin_amdgcn_wmma_f32_16x16x4_f32(false, a, false, b,
                                                  (short)0, c, false, false);
    }
    int mbase = tile * 16 + half * 8;         // C: VGPR r -> M = r + half*8
#pragma unroll
    for (int r = 0; r < 8; ++r)
        C[(mbase + r) * HID + col] = c[r];
}

// ------------- self-loop + bias init: agg = b + dinv^2 * hw (float4) --------

__global__ void k_init_agg(const float4* __restrict__ hw, const float* __restrict__ dinv,
                           const float4* __restrict__ bias4, float4* __restrict__ agg) {
    int i = blockIdx.x * blockDim.x + threadIdx.x;   // over N_NODES * (HID/4)
    if (i < N_NODES * (HID / 4)) {
        int n  = i >> 2;
        int fq = i & 3;
        float d  = dinv[n];
        float d2 = d * d;
        float4 h = hw[i];
        float4 b = bias4[fq];
        float4 o;
        o.x = b.x + d2 * h.x;
        o.y = b.y + d2 * h.y;
        o.z = b.z + d2 * h.z;
        o.w = b.w + d2 * h.w;
        agg[i] = o;
    }
}

// ---------------- edge scatter: agg[dst] += dinv[src]*dinv[dst]*hw[src] ------
// 4 threads per edge, 4 features each (float4 gather, 4 hw fp32 atomics).

__global__ void k_scatter(const int* __restrict__ ei, const float* __restrict__ dinv,
                          const float* __restrict__ hw, float* __restrict__ agg) {
    int t = blockIdx.x * blockDim.x + threadIdx.x;
    if (t >= N_EDGES * 4) return;
    int e  = t >> 2;
    int fq = (t & 3) * 4;
    int s  = ei[e];
    int d  = ei[N_EDGES + e];
    float nrm = dinv[s] * dinv[d];
    float4 v = *(const float4*)(hw + s * HID + fq);
    float* ad = agg + d * HID + fq;
    atomic_fadd_hw(ad + 0, nrm * v.x);
    atomic_fadd_hw(ad + 1, nrm * v.y);
    atomic_fadd_hw(ad + 2, nrm * v.z);
    atomic_fadd_hw(ad + 3, nrm * v.w);
}

// ---------------- final: out[i][j] = sum_k G[i][k] * hd[j][k] ----------------
// Rank-16 outer product; 8 waves per block, each wave one 16x16 tile along j.
// B fragment is hd^T, loaded directly from row-major hd (B[k][n] = hd[n][k]).

__global__ void k_outer(const float* __restrict__ G, const float* __restrict__ hd,
                        float* __restrict__ out) {
    int wave = threadIdx.x >> 5;
    int lane = threadIdx.x & 31;
    int ti = blockIdx.x;
    int tj = blockIdx.y * 8 + wave;
    if (tj >= N_DRUGS / 16) return;           // whole-wave uniform exit

    int half = lane >> 4;
    int col  = lane & 15;
    int row  = ti * 16 + (lane & 15);
    int bn   = tj * 16 + col;

    v8f c = {};
#pragma unroll
    for (int k0 = 0; k0 < HID; k0 += 4) {
        int ka = k0 + half * 2;
        v2f a, b;
        a.x = G[row * HID + ka];  a.y = G[row * HID + ka + 1];
        b.x = hd[bn * HID + ka];  b.y = hd[bn * HID + ka + 1];
        c = __builtin_amdgcn_wmma_f32_16x16x4_f32(false, a, false, b,
                                                  (short)0, c, false, false);
    }
    int mbase = ti * 16 + half * 8;
#pragma unroll
    for (int r = 0; r < 8; ++r)
        out[(mbase + r) * N_DRUGS + tj * 16 + col] = c[r];
}

// ---------------- host launch ----------------

extern "C" void kernel_launch(void* const* d_in, const int* in_sizes, int n_in,
                              void* d_out, int out_size, void* d_ws, size_t ws_size,
                              hipStream_t stream) {
    const float* x  = (const float*)d_in[0];
    const int*   ei = (const int*)  d_in[1];
    const float* W1 = (const float*)d_in[2];
    const float* b1 = (const float*)d_in[3];
    const float* W2 = (const float*)d_in[4];
    const float* b2 = (const float*)d_in[5];
    const float* P  = (const float*)d_in[6];
    float* out = (float*)d_out;

    float* ws   = (float*)d_ws;
    float* dinv = ws;  ws += N_NODES;
    float* hw1  = ws;  ws += (size_t)N_NODES * HID;
    float* agg1 = ws;  ws += (size_t)N_NODES * HID;
    float* hw2  = ws;  ws += (size_t)N_NODES * HID;
    float* agg2 = ws;  ws += (size_t)N_NODES * HID;
    float* G    = ws;  ws += (size_t)N_DRUGS * HID;

    const int ntiles_n = N_NODES / 16;   // 6250
    const int ntiles_d = N_DRUGS / 16;   // 125

    // normalization
    k_deg_init <<<(N_NODES + 255) / 256, 256, 0, stream>>>(dinv);
    k_deg_accum<<<(N_EDGES + 255) / 256, 256, 0, stream>>>(ei, dinv);
    k_rsqrt    <<<(N_NODES + 255) / 256, 256, 0, stream>>>(dinv);

    // layer 1: hw1 = x @ W1 ; agg1 = b1 + selfloop + scatter
    k_gemm16<F_IN, false><<<(ntiles_n + 7) / 8, 256, 0, stream>>>(x, W1, hw1, ntiles_n);
    k_init_agg<<<(N_NODES * (HID / 4) + 255) / 256, 256, 0, stream>>>(
        (const float4*)hw1, dinv, (const float4*)b1, (float4*)agg1);
    k_scatter <<<(N_EDGES * 4 + 255) / 256, 256, 0, stream>>>(ei, dinv, hw1, agg1);

    // layer 2: hw2 = relu(agg1) @ W2 ; agg2 = b2 + selfloop + scatter
    k_gemm16<HID, true><<<(ntiles_n + 7) / 8, 256, 0, stream>>>(agg1, W2, hw2, ntiles_n);
    k_init_agg<<<(N_NODES * (HID / 4) + 255) / 256, 256, 0, stream>>>(
        (const float4*)hw2, dinv, (const float4*)b2, (float4*)agg2);
    k_scatter <<<(N_EDGES * 4 + 255) / 256, 256, 0, stream>>>(ei, dinv, hw2, agg2);

    // predictor: G = hd @ P ; out = G @ hd^T
    k_gemm16<HID, false><<<(ntiles_d + 7) / 8, 256, 0, stream>>>(agg2, P, G, ntiles_d);
    dim3 og(ntiles_d, (ntiles_d + 7) / 8);
    k_outer<<<og, 256, 0, stream>>>(G, agg2, out);
}